// PointpairAttentionLayer_20100446945954
// MI455X (gfx1250) — compile-verified
//
#include <hip/hip_runtime.h>

typedef __attribute__((ext_vector_type(2))) float v2f;
typedef __attribute__((ext_vector_type(8))) float v8f;

// Problem constants (from reference setup_inputs)
#define BDIMC 8
#define NPTS  1024
#define KNBR  32
#define FDIM  128     // also G
#define XSS   130     // xs stride  ([k][f], padded: conflict-free f2 reads + writes)
#define ASS   130     // as stride  ([k][g])
#define OSS   33      // os stride  ([g][k])

__global__ __launch_bounds__(256)
void ppattn_fused(const float* __restrict__ x,          // (B,F,N,K)
                  const int*   __restrict__ core_types, // (B,N)
                  const int*   __restrict__ target_types,// (B,N,K)
                  const float* __restrict__ W,          // (F,F)
                  const float* __restrict__ a_pair,     // (36,F)
                  float* __restrict__ out,              // (B,G,N,K)
                  float* __restrict__ attn)             // (B,N,K,G)
{
    __shared__ float xs[KNBR * XSS];   // x tile transposed [k][f]; later reused for attention [k][g]
    __shared__ float as_[KNBR * ASS];  // gathered a_pair rows [k][g]
    __shared__ float os_[FDIM * OSS];  // elu output staging [g][k]
    __shared__ int   idxs[KNBR];

    const int bn   = blockIdx.x;          // b*N + n
    const int b    = bn >> 10;
    const int n    = bn & (NPTS - 1);
    const int t    = threadIdx.x;
    const int lane = t & 31;
    const int wave = t >> 5;

    // ---------- phase 0: stage x tile (transposed) + pair indices ----------
    {
        const int k  = lane;               // coalesced 128B per wave-instruction
        const float* xp = x + (((size_t)b * FDIM) * NPTS + n) * KNBR + k;
        #pragma unroll
        for (int i = 0; i < 16; ++i) {
            const int f = wave + (i << 3); // 8 waves x 16 iters cover f = 0..127
            xs[k * XSS + f] = xp[(size_t)f * (NPTS * KNBR)];
        }
        if (t < KNBR) {                    // wave 0 only -> no divergence
            const int c  = core_types[bn];
            const int tt = target_types[bn * KNBR + t];
            const int lo = c < tt ? c : tt;
            const int hi = c < tt ? tt : c;
            // upper-triangle pair index, C=8: start(lo) = lo*8 - lo*(lo-1)/2
            idxs[t] = lo * 8 - ((lo * (lo - 1)) >> 1) + (hi - lo);
        }
    }
    __syncthreads();

    // ---------- phase 1: gather needed a_pair rows into LDS ----------
    {
        const int k  = t >> 3;
        const int gs = (t & 7) << 4;
        const float4* ap = (const float4*)(a_pair + idxs[k] * FDIM + gs);
        #pragma unroll
        for (int j = 0; j < 4; ++j) {
            float4 v = ap[j];
            as_[k * ASS + gs + 4 * j + 0] = v.x;
            as_[k * ASS + gs + 4 * j + 1] = v.y;
            as_[k * ASS + gs + 4 * j + 2] = v.z;
            as_[k * ASS + gs + 4 * j + 3] = v.w;
        }
    }

    // ---------- GEMM: Wh[k][g] = sum_f xs[k][f] * W[f][g] via WMMA f32 16x16x4 ----------
    // wave -> g-strip [16*wave, 16*wave+16); m-tiles 0 (k=0..15) and 1 (k=16..31)
    const int m    = lane & 15;
    const int gcol = (wave << 4) + m;
    const int hk   = (lane >> 4) << 1;     // A/B K-halves: lanes 0-15 -> K=0,1 ; 16-31 -> K=2,3
    v8f acc0 = {};
    v8f acc1 = {};
    #pragma unroll 4
    for (int fb = 0; fb < FDIM; fb += 4) {
        const int fo = fb + hk;
        v2f a0 = *(const v2f*)(&xs[m * XSS + fo]);          // A frag, m-tile 0
        v2f a1 = *(const v2f*)(&xs[(m + 16) * XSS + fo]);   // A frag, m-tile 1
        v2f bf;
        bf.x = W[fo * FDIM + gcol];                         // B frag (K=fo, N=gcol)
        bf.y = W[(fo + 1) * FDIM + gcol];
        acc0 = __builtin_amdgcn_wmma_f32_16x16x4_f32(false, a0, false, bf, (short)0, acc0, false, false);
        acc1 = __builtin_amdgcn_wmma_f32_16x16x4_f32(false, a1, false, bf, (short)0, acc1, false, false);
    }
    __syncthreads();   // all xs reads done (xs reused below); as_ fully staged

    // ---------- epilogue: leaky_relu, K-softmax (lane ^ 16 partner), elu ----------
    {
        const int hb = (lane >> 4) << 3;   // C layout: VGPR r -> M = r (+8 for hi lanes)
        float wh[16], ev[16];
        float mx = -3.402823466e+38f;
        #pragma unroll
        for (int r = 0; r < 8; ++r) {
            const int k0 = hb + r;
            const int k1 = 16 + hb + r;
            float w0 = acc0[r];
            float w1 = acc1[r];
            float e0 = w0 * as_[k0 * ASS + gcol];
            float e1 = w1 * as_[k1 * ASS + gcol];
            e0 = e0 >= 0.f ? e0 : 0.2f * e0;   // leaky_relu, slope 0.2
            e1 = e1 >= 0.f ? e1 : 0.2f * e1;
            wh[r] = w0; wh[8 + r] = w1;
            ev[r] = e0; ev[8 + r] = e1;
            mx = fmaxf(mx, fmaxf(e0, e1));
        }
        mx = fmaxf(mx, __shfl_xor(mx, 16, 32));   // partner lane holds the other 16 k's
        float s = 0.f;
        #pragma unroll
        for (int i = 0; i < 16; ++i) {
            ev[i] = __expf(ev[i] - mx);
            s += ev[i];
        }
        s += __shfl_xor(s, 16, 32);
        const float rinv = 1.0f / s;
        #pragma unroll
        for (int r = 0; r < 8; ++r) {
            const int k0 = hb + r;
            const int k1 = 16 + hb + r;
            float a0v = ev[r] * rinv;
            float a1v = ev[8 + r] * rinv;
            xs[k0 * XSS + gcol] = a0v;            // stage attention [k][g] (reuse xs)
            xs[k1 * XSS + gcol] = a1v;
            float h0 = a0v * wh[r];
            float h1 = a1v * wh[8 + r];
            os_[gcol * OSS + k0] = h0 > 0.f ? h0 : (__expf(h0) - 1.0f);   // elu
            os_[gcol * OSS + k1] = h1 > 0.f ? h1 : (__expf(h1) - 1.0f);
        }
    }
    __syncthreads();

    // ---------- coalesced writes ----------
    {   // attention: (B,N,K,G) contiguous in g
        const int k  = t >> 3;
        const int gs = (t & 7) << 4;
        float4* dst = (float4*)(attn + ((size_t)bn * KNBR + k) * FDIM + gs);
        #pragma unroll
        for (int j = 0; j < 4; ++j) {
            float4 v;
            v.x = xs[k * XSS + gs + 4 * j + 0];
            v.y = xs[k * XSS + gs + 4 * j + 1];
            v.z = xs[k * XSS + gs + 4 * j + 2];
            v.w = xs[k * XSS + gs + 4 * j + 3];
            dst[j] = v;
        }
    }
    {   // out: (B,G,N,K) contiguous in k
        const int g  = t >> 1;
        const int kh = (t & 1) << 4;
        float4* dst = (float4*)(out + (((size_t)(b * FDIM + g)) * NPTS + n) * KNBR + kh);
        #pragma unroll
        for (int j = 0; j < 4; ++j) {
            float4 v;
            v.x = os_[g * OSS + kh + 4 * j + 0];
            v.y = os_[g * OSS + kh + 4 * j + 1];
            v.z = os_[g * OSS + kh + 4 * j + 2];
            v.w = os_[g * OSS + kh + 4 * j + 3];
            dst[j] = v;
        }
    }
}

extern "C" void kernel_launch(void* const* d_in, const int* in_sizes, int n_in,
                              void* d_out, int out_size, void* d_ws, size_t ws_size,
                              hipStream_t stream) {
    (void)in_sizes; (void)n_in; (void)d_ws; (void)ws_size; (void)out_size;
    const float* x          = (const float*)d_in[0];
    const int*   core_types = (const int*)d_in[1];
    const int*   target_ts  = (const int*)d_in[2];
    const float* W          = (const float*)d_in[3];
    const float* a_pair     = (const float*)d_in[4];
    // d_in[5] = lin_w, d_in[6] = lin_b : only feed the dead attention_viz branch -> unused.

    float* out  = (float*)d_out;                                      // (B,G,N,K) = 33554432 floats
    float* attn = out + (size_t)BDIMC * FDIM * NPTS * KNBR;           // (B,N,K,G) = 33554432 floats

    ppattn_fused<<<BDIMC * NPTS, 256, 0, stream>>>(x, core_types, target_ts, W, a_pair, out, attn);
}